// CoOccurrenceLayer_64811056497365
// MI455X (gfx1250) — compile-verified
//
#include <hip/hip_runtime.h>
#include <hip/hip_bf16.h>
#include <stdint.h>

typedef unsigned int u32;
typedef unsigned long long u64;
typedef u32 u32x4 __attribute__((ext_vector_type(4)));
typedef u32 u32x8 __attribute__((ext_vector_type(8)));

#define NQ    16
#define IMG   512
#define BATCH 64
#define TW    64
#define TH    8
#define TPW   (TW + 4)
#define TPH   (TH + 4)
#define EPSQ  1e-5f

// monotone float <-> ordered-uint mapping for atomic min/max
__device__ __forceinline__ u32 f2ord(float f) {
  u32 b = __float_as_uint(f);
  return (b & 0x80000000u) ? ~b : (b | 0x80000000u);
}
__device__ __forceinline__ float ord2f(u32 u) {
  u32 b = (u & 0x80000000u) ? (u & 0x7fffffffu) : ~u;
  return __uint_as_float(b);
}

__global__ void init_ws(u32* ws) {
  ws[0] = 0xFFFFFFFFu;  // min accumulator (ordered space)
  ws[1] = 0u;           // max accumulator
}

// Global min/max of x with b128 loads (HBM pass 1; warms the 192MB L2).
__global__ __launch_bounds__(256) void reduce_minmax(const float4* __restrict__ x4,
                                                     int n4, const float* __restrict__ x,
                                                     int n, u32* ws) {
  float lmin = INFINITY, lmax = -INFINITY;
  for (int i = blockIdx.x * blockDim.x + threadIdx.x; i < n4;
       i += gridDim.x * blockDim.x) {
    float4 v = x4[i];
    lmin = fminf(lmin, fminf(fminf(v.x, v.y), fminf(v.z, v.w)));
    lmax = fmaxf(lmax, fmaxf(fmaxf(v.x, v.y), fmaxf(v.z, v.w)));
  }
  // scalar tail (n not divisible by 4), covered by block 0
  if (blockIdx.x == 0) {
    for (int i = n4 * 4 + threadIdx.x; i < n; i += blockDim.x) {
      float v = x[i];
      lmin = fminf(lmin, v);
      lmax = fmaxf(lmax, v);
    }
  }
#pragma unroll
  for (int off = 16; off > 0; off >>= 1) {   // wave32 butterfly reduce
    lmin = fminf(lmin, __shfl_xor(lmin, off, 32));
    lmax = fmaxf(lmax, __shfl_xor(lmax, off, 32));
  }
  if ((threadIdx.x & 31) == 0) {
    atomicMin(&ws[0], f2ord(lmin));
    atomicMax(&ws[1], f2ord(lmax));
  }
}

// Fused bin + data-dependent 5x5 stencil.
// out[p] = sum_d filt[d] * C[bin(p), bin(p+d)] * x[p+d]  (zero-padded)
__global__ __launch_bounds__(256) void cooc_kernel(const float* __restrict__ x,
                                                   const float* __restrict__ co,
                                                   const float* __restrict__ filt,
                                                   float* __restrict__ out,
                                                   const u32* __restrict__ ws) {
  __shared__ float  xs[TPH * TPW];   // staged x tile (+halo), filled by TDM
  __shared__ float2 pk[TPH * TPW];   // packed {x, as_float(bin)} per staged pixel
  __shared__ float  Ctab[NQ * NQ];   // co-occurrence table

  const int tid = threadIdx.x;
  const int c0  = blockIdx.x * TW;
  const int r0  = blockIdx.y * TH;
  const int b   = blockIdx.z;

  // quantizer scales from global min/max
  const float emin = expf(ord2f(ws[0]));
  const float emax = expf(ord2f(ws[1]));
  const float s1 = (float)NQ / emax;
  const float s2 = emin * s1 + EPSQ;

  // staged rectangle, clamped to the flat [64*512, 512] buffer (no OOB reads).
  const int gr0  = b * IMG + r0;            // flat row of tile origin
  const int srow = max(gr0 - 2, 0);
  const int cutT = srow - (gr0 - 2);        // 0 or 2 (only b==0,r0==0)
  const int scol = max(c0 - 2, 0);
  const int cutL = scol - (c0 - 2);         // 0 or 2 (only c0==0)
  const int width = min(TPW - cutL, IMG - scol);
  const int rows  = min(TPH - cutT, BATCH * IMG - srow);

  if (tid < NQ * NQ) Ctab[tid] = co[tid];

  // -------- Tensor Data Mover: one 2D strided tile -> LDS (wave 0 issues) ----
  if (tid < 32) {
    u64 ga  = (u64)(uintptr_t)x + ((u64)srow * IMG + (u64)scol) * 4ull;
    u32 lds = (u32)(uintptr_t)(void*)&xs[0];  // low 32 bits = LDS byte offset

    u32x4 g0, g2, g3;
    u32x8 g1;
    // D# group 0: count=1 | lds_addr | global_addr(57b) | type=2
    g0[0] = 1u;
    g0[1] = lds;
    g0[2] = (u32)(ga & 0xFFFFFFFFull);
    g0[3] = (u32)((ga >> 32) & 0x01FFFFFFull) | 0x80000000u;
    // D# group 1: mask=0, data_size=4B, tensor 512 x 32768 stride 512,
    //             tile = width x rows
    g1[0] = 2u << 16;                                   // data_size = 4B
    g1[1] = ((u32)IMG & 0xFFFFu) << 16;                 // tensor_dim0[15:0]
    g1[2] = ((u32)IMG >> 16) | (((u32)(BATCH * IMG) & 0xFFFFu) << 16);
    g1[3] = ((u32)(BATCH * IMG) >> 16) | ((u32)width << 16);  // tile_dim0
    g1[4] = (u32)rows;                                  // tile_dim1, tile_dim2=0
    g1[5] = (u32)IMG;                                   // tensor_dim0_stride
    g1[6] = 0u;
    g1[7] = 0u;
    g2[0] = 1u; g2[1] = 0u; g2[2] = 0u; g2[3] = 0u;     // tensor_dim2=1 (unused)
    g3[0] = 0u; g3[1] = 0u; g3[2] = 0u; g3[3] = 0u;
    asm volatile("tensor_load_to_lds %0, %1, %2, %3"
                 :: "s"(g0), "s"(g1), "s"(g2), "s"(g3)
                 : "memory");
    __builtin_amdgcn_s_wait_tensorcnt(0);
  }
  __syncthreads();

  // bin every staged pixel once, pack {x, bin} for single-b64 taps later
  for (int i = tid; i < rows * width; i += 256) {
    float v  = xs[i];
    int   bi = (int)floorf(fabsf(expf(v) * s1 - s2));
    bi = min(max(bi, 0), NQ - 1);
    pk[i] = make_float2(v, __int_as_float(bi));
  }
  __syncthreads();

  // 25 filter taps -> registers (uniform address -> scalar loads)
  float fw[25];
#pragma unroll
  for (int k = 0; k < 25; ++k) fw[k] = filt[k];

  const int tx  = tid & (TW - 1);
  const int ty0 = (tid >> 6) * 2;

#pragma unroll
  for (int sub = 0; sub < 2; ++sub) {
    const int ty = ty0 + sub;
    const int y  = r0 + ty;
    const int xc = c0 + tx;
    const int ci = __float_as_int(pk[(ty + 2 - cutT) * width + (tx + 2 - cutL)].y);
    const float* Crow = &Ctab[ci * NQ];
    float acc = 0.0f;
#pragma unroll
    for (int dy = 0; dy < 5; ++dy) {
      const int ny  = y + dy - 2;
      const bool rok = (ny >= 0) && (ny < IMG);
      const int lr  = (ty + dy - cutT) * width;
#pragma unroll
      for (int dx = 0; dx < 5; ++dx) {
        const int nx  = xc + dx - 2;
        const bool ok = rok && (nx >= 0) && (nx < IMG);
        int li = lr + (tx + dx - cutL);
        li = ok ? li : 0;                      // safe staged index
        const float2 p = pk[li];               // one ds_load_b64 per tap
        const float  w = Crow[__float_as_int(p.y)];
        acc += ok ? (fw[dy * 5 + dx] * w * p.x) : 0.0f;
      }
    }
    // NT store: stream out past L2 so x stays L2-resident (67MB < 192MB L2)
    __builtin_nontemporal_store(acc, &out[((u64)b * IMG + y) * IMG + xc]);
  }
}

extern "C" void kernel_launch(void* const* d_in, const int* in_sizes, int n_in,
                              void* d_out, int out_size, void* d_ws, size_t ws_size,
                              hipStream_t stream) {
  (void)n_in; (void)out_size; (void)ws_size;
  const float* x    = (const float*)d_in[0];
  const float* co   = (const float*)d_in[1];
  const float* filt = (const float*)d_in[2];
  float* out = (float*)d_out;
  u32*   ws  = (u32*)d_ws;
  const int n  = in_sizes[0];  // 64*1*512*512
  const int n4 = n >> 2;

  init_ws<<<1, 1, 0, stream>>>(ws);
  reduce_minmax<<<2048, 256, 0, stream>>>((const float4*)x, n4, x, n, ws);
  dim3 grid(IMG / TW, IMG / TH, BATCH);  // 8 x 64 x 64 tiles
  cooc_kernel<<<grid, 256, 0, stream>>>(x, co, filt, out, ws);
}